// TransformerEncoderLayer_37709812859299
// MI455X (gfx1250) — compile-verified
//
#include <hip/hip_runtime.h>
#include <hip/hip_bf16.h>
#include <math.h>

// ---------------------------------------------------------------------------
// Multi-scale deformable attention encoder layer for gfx1250 (MI455X).
// All GEMMs run on the CDNA5 matrix pipe via V_WMMA_F32_16X16X4_F32
// (fp32 in/out, bit-compatible with the fp32 JAX reference).
// Weights are repacked once per launch into a row-pair-interleaved layout so
// every B fragment is a single 8-byte global_load_b64 (the round-1 assembly
// showed the B path issuing 2x global_load_b32 per fragment and dominating
// instruction issue in the WMMA loop).
// Token ordering everywhere: [b][level][pos][c]  ==  flat d_out layout.
// ---------------------------------------------------------------------------

typedef __attribute__((ext_vector_type(2))) float v2f;
typedef __attribute__((ext_vector_type(8))) float v8f;

#define NTOK_B   5440      // tokens per batch (64*64 + 32*32 + 16*16 + 8*8)
#define NTOK     10880     // B * NTOK_B
#define DMODEL   256
#define DFF      2048
#define NHEADS   8
#define DK       32

// ---------------------------------------------------------------------------
// Repack W[K,N] (row-major) -> Wp[K/2][N][2] so that the two B-fragment
// elements a lane needs per k-step (rows k, k+1 at one column) are adjacent.
// ---------------------------------------------------------------------------
__global__ __launch_bounds__(256) void repack_w_kernel(
    const float* __restrict__ W, float* __restrict__ Wp, int K, int N)
{
  const int o = blockIdx.x * 256 + threadIdx.x;
  if (o >= K * N) return;
  const int kp  = o / (2 * N);
  const int rem = o - kp * 2 * N;
  const int n   = rem >> 1;
  const int r   = rem & 1;
  Wp[o] = W[(size_t)(2 * kp + r) * N + n];
}

// ---------------------------------------------------------------------------
// Generic fp32 WMMA GEMM:  Y[M,N] (+)= X[M,K] @ W[K,N] (+ bias) (relu?)
// W is pre-packed (see repack_w_kernel); Np = packed row stride (original N
// of the full matrix; column slices are taken by pre-offsetting the pointer).
// One wave computes a 16x64 tile: A-fragment reused across 4 WMMA per k-step.
// Fragment layouts per CDNA5 ISA 7.12.2 (f32 16x16x4):
//   A: VGPR j = K=j+2*half, lanes 0..15 = M rows (half = lane>=16)
//   B: VGPR j = K=j+2*half row, N = lane%16
//   C/D: VGPR j -> M = j + 8*half, N = lane%16
// ---------------------------------------------------------------------------
__global__ __launch_bounds__(256) void wmma_gemm_f32(
    const float* __restrict__ X, int ldx,
    const float* __restrict__ Wp, int Np,
    const float* __restrict__ bias,
    float* __restrict__ Y, int ldy,
    int M, int K, int ntiles64, int relu, int accum)
{
  const int wave = blockIdx.x * 8 + (threadIdx.x >> 5);
  const int mtiles = M >> 4;
  const int tileM = wave / ntiles64;
  const int tileN = wave - tileM * ntiles64;
  if (tileM >= mtiles) return;          // wave-uniform: EXEC all-1 inside

  const int lane = threadIdx.x & 31;
  const int half = lane >> 4;           // 0: K+{0,1} | 1: K+{2,3}
  const int lr   = lane & 15;

  // A fragment source: v2f elements {X[r][k+2h], X[r][k+2h+1]}
  const v2f* xv = (const v2f*)(X + (size_t)(tileM * 16 + lr) * ldx + 2 * half);
  // B fragment source: packed pairs; pair-row index = k/2 + half
  const v2f* wvb = (const v2f*)Wp + (size_t)half * Np + tileN * 64 + lr;

  v8f acc0 = {}, acc1 = {}, acc2 = {}, acc3 = {};

  for (int k = 0; k < K; k += 4) {
    const v2f a = xv[k >> 1];
    const v2f* wk = wvb + (size_t)(k >> 1) * Np;
    const v2f b0 = wk[0];
    const v2f b1 = wk[16];
    const v2f b2 = wk[32];
    const v2f b3 = wk[48];
    acc0 = __builtin_amdgcn_wmma_f32_16x16x4_f32(false, a, false, b0, (short)0, acc0, false, false);
    acc1 = __builtin_amdgcn_wmma_f32_16x16x4_f32(false, a, false, b1, (short)0, acc1, false, false);
    acc2 = __builtin_amdgcn_wmma_f32_16x16x4_f32(false, a, false, b2, (short)0, acc2, false, false);
    acc3 = __builtin_amdgcn_wmma_f32_16x16x4_f32(false, a, false, b3, (short)0, acc3, false, false);
  }

  v8f accs[4] = {acc0, acc1, acc2, acc3};
#pragma unroll
  for (int n = 0; n < 4; ++n) {
    const int col = tileN * 64 + n * 16 + lr;
    const float bv = bias ? bias[col] : 0.0f;
#pragma unroll
    for (int j = 0; j < 8; ++j) {
      float v = accs[n][j] + bv;
      if (relu) v = fmaxf(v, 0.0f);
      const size_t oi = (size_t)(tileM * 16 + j + 8 * half) * ldy + col;
      if (accum) Y[oi] += v;
      else       Y[oi] = v;
    }
  }
}

// ---------------------------------------------------------------------------
// PE = src + pos, gathered from the 4 per-level input tensors into the
// flat [b][level][pos][c] layout.
// ---------------------------------------------------------------------------
__global__ __launch_bounds__(256) void add_pe_kernel(
    const float* __restrict__ s0, const float* __restrict__ p0,
    const float* __restrict__ s1, const float* __restrict__ p1,
    const float* __restrict__ s2, const float* __restrict__ p2,
    const float* __restrict__ s3, const float* __restrict__ p3,
    float* __restrict__ PE)
{
  const int e = blockIdx.x * 256 + threadIdx.x;   // [0, NTOK*256)
  const int t = e >> 8, c = e & 255;
  const int b = t / NTOK_B;
  const int r = t - b * NTOK_B;
  const float* sp; const float* pp; int pos, nl;
  if (r < 4096)      { sp = s0; pp = p0; pos = r;        nl = 4096; }
  else if (r < 5120) { sp = s1; pp = p1; pos = r - 4096; nl = 1024; }
  else if (r < 5376) { sp = s2; pp = p2; pos = r - 5120; nl = 256;  }
  else               { sp = s3; pp = p3; pos = r - 5376; nl = 64;   }
  const size_t idx = ((size_t)b * nl + pos) * 256 + c;
  PE[e] = sp[idx] + pp[idx];
}

// ---------------------------------------------------------------------------
// In-place softmax over the 16 (L*K) attention logits per (token, head).
// ---------------------------------------------------------------------------
__global__ __launch_bounds__(256) void softmax16_kernel(float* __restrict__ A)
{
  const int i = blockIdx.x * 256 + threadIdx.x;   // [0, NTOK*8)
  if (i >= NTOK * NHEADS) return;
  const int t = i >> 3, m = i & 7;
  float* p = A + (size_t)t * 128 + m * 16;
  float mx = -3.0e38f;
#pragma unroll
  for (int j = 0; j < 16; ++j) mx = fmaxf(mx, p[j]);
  float e[16]; float s = 0.0f;
#pragma unroll
  for (int j = 0; j < 16; ++j) { e[j] = expf(p[j] - mx); s += e[j]; }
  const float inv = 1.0f / s;
#pragma unroll
  for (int j = 0; j < 16; ++j) p[j] = e[j] * inv;
}

// ---------------------------------------------------------------------------
// Deformable sampling: one wave per (token, head); 32 lanes = 32 channels so
// every bilinear corner gather is one coalesced 128B read of head-major V.
// Zero padding outside [0, w-1] (align_corners pixel coords, matching ref).
// ---------------------------------------------------------------------------
__global__ __launch_bounds__(256) void msda_sample_kernel(
    const float* __restrict__ V,    // [NTOK,256] head-major value proj
    const float* __restrict__ OFF,  // [NTOK,256] = [m][l][k][xy]
    const float* __restrict__ A,    // [NTOK,128] = [m][l*K+k], softmaxed
    const float* __restrict__ r0, const float* __restrict__ r1,
    const float* __restrict__ r2, const float* __restrict__ r3,
    float* __restrict__ ACC)        // [NTOK,256]
{
  const int w = blockIdx.x * 8 + (threadIdx.x >> 5);
  const int t = w >> 3, m = w & 7;
  if (t >= NTOK) return;
  const int lane = threadIdx.x & 31;

  const int b = t / NTOK_B;
  const int r = t - b * NTOK_B;
  const float* rp; int pos, nq;
  if (r < 4096)      { rp = r0; pos = r;        nq = 4096; }
  else if (r < 5120) { rp = r1; pos = r - 4096; nq = 1024; }
  else if (r < 5376) { rp = r2; pos = r - 5120; nq = 256;  }
  else               { rp = r3; pos = r - 5376; nq = 64;   }
  const float rx = rp[((size_t)b * nq + pos) * 2 + 0];
  const float ry = rp[((size_t)b * nq + pos) * 2 + 1];

  const int LSZ[4]  = {64, 32, 16, 8};
  const int LOFF[4] = {0, 4096, 5120, 5376};

  const float* offp = OFF + (size_t)t * 256 + m * 32;  // ((m*4+l)*4+k)*2+d
  const float* ap   = A   + (size_t)t * 128 + m * 16;

  float acc = 0.0f;
  const size_t vb = (size_t)b * NTOK_B;
#pragma unroll
  for (int l = 0; l < 4; ++l) {
    const int wl = LSZ[l];
    const float sx = rx * (float)(wl - 1);
    const float sy = ry * (float)(wl - 1);
    const size_t vbase = (vb + LOFF[l]) * (size_t)256 + m * DK + lane;
#pragma unroll
    for (int k = 0; k < 4; ++k) {
      const float px = sx + offp[(l * 4 + k) * 2 + 0];
      const float py = sy + offp[(l * 4 + k) * 2 + 1];
      const float aw = ap[l * 4 + k];
      const float xf = floorf(px), yf = floorf(py);
      const float fx = px - xf, fy = py - yf;
      const int x0 = (int)xf, y0 = (int)yf;
      float sacc = 0.0f;
#pragma unroll
      for (int dy = 0; dy < 2; ++dy) {
        const int yy = y0 + dy;
        const float wy = dy ? fy : 1.0f - fy;
        if (yy < 0 || yy >= wl) continue;
#pragma unroll
        for (int dx = 0; dx < 2; ++dx) {
          const int xx = x0 + dx;
          const float wx = dx ? fx : 1.0f - fx;
          if (xx < 0 || xx >= wl) continue;
          sacc += wy * wx * V[vbase + ((size_t)yy * wl + xx) * 256];
        }
      }
      acc += aw * sacc;
    }
  }
  ACC[(size_t)t * 256 + m * DK + lane] = acc;
}

// ---------------------------------------------------------------------------
// out = LayerNorm(A + Bres (+ cbias)) * g + be, one wave per token (wave32).
// ---------------------------------------------------------------------------
__global__ __launch_bounds__(256) void ln_res_kernel(
    const float* __restrict__ A, const float* __restrict__ Bres,
    const float* __restrict__ cbias,    // nullable per-channel bias (bf2)
    const float* __restrict__ g, const float* __restrict__ be,
    float* __restrict__ out, int T)
{
  const int t = blockIdx.x * 8 + (threadIdx.x >> 5);
  if (t >= T) return;
  const int lane = threadIdx.x & 31;
  const float* a = A + (size_t)t * 256;
  const float* b = Bres + (size_t)t * 256;

  float v[8]; float s = 0.0f, s2 = 0.0f;
#pragma unroll
  for (int i = 0; i < 8; ++i) {
    const int ch = lane + 32 * i;
    float x = a[ch] + b[ch];
    if (cbias) x += cbias[ch];
    v[i] = x; s += x; s2 += x * x;
  }
#pragma unroll
  for (int off = 16; off >= 1; off >>= 1) {
    s  += __shfl_xor(s,  off, 32);
    s2 += __shfl_xor(s2, off, 32);
  }
  const float mu  = s * (1.0f / 256.0f);
  const float var = s2 * (1.0f / 256.0f) - mu * mu;
  const float rst = rsqrtf(var + 1e-5f);
#pragma unroll
  for (int i = 0; i < 8; ++i) {
    const int ch = lane + 32 * i;
    out[(size_t)t * 256 + ch] = (v[i] - mu) * rst * g[ch] + be[ch];
  }
}

// ---------------------------------------------------------------------------
// Host-side orchestration (all launches on `stream`; graph-capture safe).
// ---------------------------------------------------------------------------
extern "C" void kernel_launch(void* const* d_in, const int* in_sizes, int n_in,
                              void* d_out, int out_size, void* d_ws, size_t ws_size,
                              hipStream_t stream) {
  (void)in_sizes; (void)n_in; (void)out_size; (void)ws_size;
  // setup_inputs() dict order: src_i,pos_i,ref_i interleaved, then weights.
  const float* src[4] = {(const float*)d_in[0], (const float*)d_in[3],
                         (const float*)d_in[6], (const float*)d_in[9]};
  const float* pos[4] = {(const float*)d_in[1], (const float*)d_in[4],
                         (const float*)d_in[7], (const float*)d_in[10]};
  const float* ref[4] = {(const float*)d_in[2], (const float*)d_in[5],
                         (const float*)d_in[8], (const float*)d_in[11]};
  const float* Wq   = (const float*)d_in[12]; const float* bq   = (const float*)d_in[13];
  const float* Wk   = (const float*)d_in[14]; const float* bk   = (const float*)d_in[15];
  const float* Woff = (const float*)d_in[16]; const float* boff = (const float*)d_in[17];
  const float* WA   = (const float*)d_in[18]; const float* bA   = (const float*)d_in[19];
  const float* Wo   = (const float*)d_in[20]; const float* bo   = (const float*)d_in[21];
  const float* W1   = (const float*)d_in[22]; const float* bf1  = (const float*)d_in[23];
  const float* W2   = (const float*)d_in[24]; const float* bf2  = (const float*)d_in[25];
  const float* g1   = (const float*)d_in[26]; const float* be1  = (const float*)d_in[27];
  const float* g2   = (const float*)d_in[28]; const float* be2  = (const float*)d_in[29];
  float* out = (float*)d_out;

  // Workspace layout (fp32), ~78 MB total, with lifetime-safe aliasing.
  float* PE  = (float*)d_ws;               // [NTOK,256] pos-embedded sources
  float* V   = PE  + (size_t)NTOK * 256;   // value proj; later aliased as S2
  float* QP  = V   + (size_t)NTOK * 256;   // query proj; later aliased as Y
  float* OFF = QP  + (size_t)NTOK * 256;   // offsets; later aliased as H-chunk
  float* Aw  = OFF + (size_t)NTOK * 256;   // [NTOK,128] attention weights
  float* ACC = Aw  + (size_t)NTOK * 128;   // sampled+weighted heads
  float* Xb  = ACC + (size_t)NTOK * 256;   // post-LN1 activations
  // packed weights (row-pair interleaved)
  float* Wqp   = Xb + (size_t)NTOK * 256;  // 256*256
  float* Wkp   = Wqp   + 256 * 256;
  float* Woffp = Wkp   + 256 * 256;
  float* WAp   = Woffp + 256 * 256;        // 256*128
  float* Wop   = WAp   + 256 * 128;
  float* W1p   = Wop   + 256 * 256;        // 256*2048
  float* W2p   = W1p   + 256 * 2048;       // 2048*256
  // aliases (lifetimes disjoint with originals)
  float* S2 = V;    // attention output (V dead after sampling)
  float* Hc = OFF;  // FFN hidden chunk (OFF dead after sampling)
  float* Yb = QP;   // FFN output accumulator (QP dead after OFF/A GEMMs)

  const int gemmBlk256 = (NTOK / 16) * (DMODEL / 64) / 8;   // 340
  const int gemmBlk128 = (NTOK / 16) * (128 / 64) / 8;      // 170

  // 0. pack all weight matrices for b64 B-fragment loads
  repack_w_kernel<<<256, 256, 0, stream>>>(Wq, Wqp, 256, 256);
  repack_w_kernel<<<256, 256, 0, stream>>>(Wk, Wkp, 256, 256);
  repack_w_kernel<<<256, 256, 0, stream>>>(Woff, Woffp, 256, 256);
  repack_w_kernel<<<128, 256, 0, stream>>>(WA, WAp, 256, 128);
  repack_w_kernel<<<256, 256, 0, stream>>>(Wo, Wop, 256, 256);
  repack_w_kernel<<<2048, 256, 0, stream>>>(W1, W1p, 256, 2048);
  repack_w_kernel<<<2048, 256, 0, stream>>>(W2, W2p, 2048, 256);

  // 1. PE = src + pos
  add_pe_kernel<<<NTOK, 256, 0, stream>>>(src[0], pos[0], src[1], pos[1],
                                          src[2], pos[2], src[3], pos[3], PE);
  // 2. V = PE @ Wk + bk   (hoisted: reference recomputes this per query level)
  wmma_gemm_f32<<<gemmBlk256, 256, 0, stream>>>(PE, 256, Wkp, 256, bk, V, 256,
                                                NTOK, 256, 4, 0, 0);
  // 3. QP = PE @ Wq + bq
  wmma_gemm_f32<<<gemmBlk256, 256, 0, stream>>>(PE, 256, Wqp, 256, bq, QP, 256,
                                                NTOK, 256, 4, 0, 0);
  // 4. OFF = QP @ Woff + boff
  wmma_gemm_f32<<<gemmBlk256, 256, 0, stream>>>(QP, 256, Woffp, 256, boff, OFF, 256,
                                                NTOK, 256, 4, 0, 0);
  // 5. Aw = QP @ WA + bA
  wmma_gemm_f32<<<gemmBlk128, 256, 0, stream>>>(QP, 256, WAp, 128, bA, Aw, 128,
                                                NTOK, 256, 2, 0, 0);
  // 6. softmax over L*K per head (in place)
  softmax16_kernel<<<(NTOK * NHEADS) / 256, 256, 0, stream>>>(Aw);
  // 7. deformable bilinear sampling -> ACC
  msda_sample_kernel<<<NTOK, 256, 0, stream>>>(V, OFF, Aw, ref[0], ref[1],
                                               ref[2], ref[3], ACC);
  // 8. S2 = ACC @ Wo + bo
  wmma_gemm_f32<<<gemmBlk256, 256, 0, stream>>>(ACC, 256, Wop, 256, bo, S2, 256,
                                                NTOK, 256, 4, 0, 0);
  // 9. X = LN1(PE + S2)
  ln_res_kernel<<<NTOK / 8, 256, 0, stream>>>(PE, S2, nullptr, g1, be1, Xb, NTOK);
  // 10. FFN in 8 column chunks of 256 (hidden never materialized in full):
  //     Hc = relu(X @ W1[:,c] + bf1[c]);  Y (+)= Hc @ W2[c,:]
  for (int c = 0; c < DFF / 256; ++c) {
    wmma_gemm_f32<<<gemmBlk256, 256, 0, stream>>>(
        Xb, 256, W1p + (size_t)c * 256 * 2, DFF, bf1 + c * 256, Hc, 256,
        NTOK, 256, 4, /*relu=*/1, /*accum=*/0);
    wmma_gemm_f32<<<gemmBlk256, 256, 0, stream>>>(
        Hc, 256, W2p + (size_t)c * 128 * 256 * 2, 256, nullptr, Yb, 256,
        NTOK, 256, 4, /*relu=*/0, /*accum=*/(c > 0));
  }
  // 11. out = LN2(X + Y + bf2)  -> d_out is exactly [b][level][pos][c] flat
  ln_res_kernel<<<NTOK / 8, 256, 0, stream>>>(Xb, Yb, bf2, g2, be2, out, NTOK);
}